// GCN_90013924590233
// MI455X (gfx1250) — compile-verified
//
#include <hip/hip_runtime.h>
#include <hip/hip_bf16.h>

typedef __attribute__((ext_vector_type(2))) float v2f;
typedef __attribute__((ext_vector_type(8))) float v8f;

#define N_NODES 100000

// ---------------------------------------------------------------------------
// init: deg = 1 (self loop). Aggregation buffers are seeded by the GEMM
// epilogues, so no big zero-fills needed.
// ---------------------------------------------------------------------------
__global__ void gcn_init_kernel(float* __restrict__ deg, int n_nodes) {
  const int i = blockIdx.x * blockDim.x + threadIdx.x;
  if (i < n_nodes) deg[i] = 1.0f;
}

// in-degree accumulation over edges (dst side, matches reference segment_sum)
__global__ void gcn_degree_kernel(const int* __restrict__ dst,
                                  float* __restrict__ deg, int n_edges) {
  const int e = blockIdx.x * blockDim.x + threadIdx.x;
  if (e >= n_edges) return;
  atomicAdd(&deg[dst[e]], 1.0f);
}

// deg -> dinv = rsqrt(deg)  (deg >= 1 always, self-loops included)
__global__ void gcn_rsqrt_kernel(float* __restrict__ deg, int n_nodes) {
  const int i = blockIdx.x * blockDim.x + threadIdx.x;
  if (i >= n_nodes) return;
  deg[i] = __frsqrt_rn(deg[i]);
}

// ---------------------------------------------------------------------------
// WMMA fp32 GEMM with fused epilogue:
//   H[M,NOUT] = A[M,KDIM] @ B[KDIM,NOUT]                    (raw features)
//   O[M,NOUT] = H * dinv[row]^2 + bias[col]                 (self-loop seed)
// Each wave computes MT stacked 16-row tiles x NOUT cols, stepping K by 4
// with V_WMMA_F32_16X16X4_F32; B fragments are reused across the MT tiles.
//
// A-frag (16x4 f32, ISA 7.12.2): lane<16 holds (M=lane, K=k0..k0+1),
//                                lane>=16 holds (M=lane-16, K=k0+2..k0+3).
// B-frag (4x16) mirrored over N. C/D: VGPR r holds M = r + 8*(lane>=16),
// N = (lane&15) within each 16-col tile.
// ---------------------------------------------------------------------------
template <int KDIM, int NOUT, int MT>
__global__ void gcn_gemm_wmma_kernel(const float* __restrict__ A,
                                     const float* __restrict__ B,
                                     float* __restrict__ H,
                                     float* __restrict__ O,
                                     const float* __restrict__ dinv,
                                     const float* __restrict__ bias, int M) {
  constexpr int NT = NOUT / 16;
  const int lane  = threadIdx.x & 31;
  const int wave  = threadIdx.x >> 5;
  const int group = blockIdx.x * (blockDim.x >> 5) + wave;  // MT*16-row group
  const int row0  = group * (16 * MT);
  if (row0 >= M) return;  // wave-uniform guard -> EXEC all-ones inside WMMA

  const int mrow = lane & 15;
  const int koff = (lane >> 4) << 1;  // 0 for lanes 0-15, 2 for lanes 16-31

  v8f acc[MT][NT] = {};
  const float* arow[MT];
#pragma unroll
  for (int m = 0; m < MT; ++m)
    arow[m] = A + (size_t)(row0 + m * 16 + mrow) * KDIM;

  for (int k0 = 0; k0 < KDIM; k0 += 4) {
    v2f a[MT];
#pragma unroll
    for (int m = 0; m < MT; ++m) {
      a[m].x = arow[m][k0 + koff];
      a[m].y = arow[m][k0 + koff + 1];
    }
#pragma unroll
    for (int t = 0; t < NT; ++t) {
      const int col = t * 16 + mrow;
      v2f b;
      b.x = B[(size_t)(k0 + koff) * NOUT + col];
      b.y = B[(size_t)(k0 + koff + 1) * NOUT + col];
#pragma unroll
      for (int m = 0; m < MT; ++m) {
        acc[m][t] = __builtin_amdgcn_wmma_f32_16x16x4_f32(
            /*neg_a=*/false, a[m], /*neg_b=*/false, b,
            /*c_mod=*/(short)0, acc[m][t], /*reuse_a=*/false, /*reuse_b=*/false);
      }
    }
  }

  // fused epilogue: write raw features + self-loop-seeded aggregation buffer
  float bs[NT];
#pragma unroll
  for (int t = 0; t < NT; ++t) bs[t] = bias[t * 16 + (lane & 15)];

  const int mbase = (lane >> 4) * 8;
#pragma unroll
  for (int m = 0; m < MT; ++m) {
#pragma unroll
    for (int r = 0; r < 8; ++r) {
      const int row = row0 + m * 16 + mbase + r;
      const float di = dinv[row];
      const float d2 = di * di;
#pragma unroll
      for (int t = 0; t < NT; ++t) {
        const float v = acc[m][t][r];
        const size_t idx = (size_t)row * NOUT + t * 16 + (lane & 15);
        H[idx] = v;
        O[idx] = v * d2 + bs[t];
      }
    }
  }
}

// ---------------------------------------------------------------------------
// Edge scatter: LPE lanes per edge, each lane moves one float4 slice.
// out[dst] += H[src] * dinv[src] * dinv[dst]  (per-dword global_atomic_add_f32)
// Prefetch the destination row toward GL2 before the atomics (atomics execute
// at L2 on random 256B rows -> this is the latency that matters).
// ---------------------------------------------------------------------------
template <int F, int LPE>
__global__ void gcn_scatter_kernel(const float* __restrict__ H,
                                   const float* __restrict__ dinv,
                                   const int* __restrict__ src,
                                   const int* __restrict__ dst,
                                   float* __restrict__ out, int n_edges) {
  const unsigned gid = blockIdx.x * blockDim.x + threadIdx.x;
  const unsigned e   = gid / LPE;
  const unsigned sub = gid % LPE;
  if (e >= (unsigned)n_edges) return;
  const int s = src[e];
  const int d = dst[e];
  float* o = out + (size_t)d * F + sub * 4;
  __builtin_prefetch(o, 0, 1);  // global_prefetch_b8 toward GL2
  const float nrm = dinv[s] * dinv[d];
  const float4 v = *(const float4*)(H + (size_t)s * F + sub * 4);
  atomicAdd(o + 0, v.x * nrm);
  atomicAdd(o + 1, v.y * nrm);
  atomicAdd(o + 2, v.z * nrm);
  atomicAdd(o + 3, v.w * nrm);
}

// ---------------------------------------------------------------------------
// Layer-1 activation, in place: agg := dropout(relu(agg))   (p=0.2, keep 0.8)
// Deterministic counter-hash keep mask (JAX PRNG not reproducible bit-exact).
// ---------------------------------------------------------------------------
__global__ void gcn_act1_kernel(float* __restrict__ agg, int total) {
  const int i = blockIdx.x * blockDim.x + threadIdx.x;
  if (i >= total) return;
  float v = agg[i];
  v = v > 0.0f ? v : 0.0f;
  unsigned x = (unsigned)i * 2654435761u + 0x9E3779B9u;
  x ^= x >> 16; x *= 0x85EBCA6Bu;
  x ^= x >> 13; x *= 0xC2B2AE35u;
  x ^= x >> 16;
  const bool keep = (x >> 8) < (unsigned)(0.8 * 16777216.0);
  agg[i] = keep ? v * 1.25f : 0.0f;
}

// ---------------------------------------------------------------------------
extern "C" void kernel_launch(void* const* d_in, const int* in_sizes, int n_in,
                              void* d_out, int out_size, void* d_ws, size_t ws_size,
                              hipStream_t stream) {
  const float* x   = (const float*)d_in[0];
  const int*   ei  = (const int*)d_in[1];     // [2, E] int32
  const float* W1  = (const float*)d_in[2];   // [128, 64]
  const float* b1  = (const float*)d_in[3];   // [64]
  const float* W2  = (const float*)d_in[4];   // [64, 32]
  const float* b2  = (const float*)d_in[5];   // [32]
  float*       out = (float*)d_out;           // [N, 32]

  const int N = N_NODES;
  const int E = in_sizes[1] / 2;
  const int* src = ei;
  const int* dst = ei + E;

  // workspace layout (floats)
  float* ws   = (float*)d_ws;
  float* dinv = ws;                     // N  (deg, then rsqrt in place)
  float* bufA = ws + 102400;            // N*64 : h1, later reused for h2 (N*32)
  float* bufB = bufA + (size_t)N * 64;  // N*64 : agg1 seed -> activated h1'

  const int TB = 256;

  // 1) deg = 1 (self loops)
  gcn_init_kernel<<<(N + TB - 1) / TB, TB, 0, stream>>>(dinv, N);

  // 2) degree accumulation + rsqrt
  gcn_degree_kernel<<<(E + TB - 1) / TB, TB, 0, stream>>>(dst, dinv, E);
  gcn_rsqrt_kernel<<<(N + TB - 1) / TB, TB, 0, stream>>>(dinv, N);

  // 3) h1 = x @ W1 ; bufB seeded with h1*dinv^2 + b1 (self-loop + bias)
  {
    const int groups = (N + 31) / 32;           // MT=2 -> 32 rows per wave
    const int blocks = (groups + 7) / 8;        // 8 waves per block
    gcn_gemm_wmma_kernel<128, 64, 2><<<blocks, TB, 0, stream>>>(
        x, W1, bufA, bufB, dinv, b1, N);
  }

  // 4) layer-1 edge scatter into bufB, then in-place relu+dropout
  gcn_scatter_kernel<64, 16><<<((long long)E * 16 + TB - 1) / TB, TB, 0, stream>>>(
      bufA, dinv, src, dst, bufB, E);
  gcn_act1_kernel<<<(N * 64 + TB - 1) / TB, TB, 0, stream>>>(bufB, N * 64);

  // 5) h2 = h1' @ W2 -> bufA ; d_out seeded with h2*dinv^2 + b2
  {
    const int groups = (N + 31) / 32;
    const int blocks = (groups + 7) / 8;
    gcn_gemm_wmma_kernel<64, 32, 2><<<blocks, TB, 0, stream>>>(
        bufB, W2, bufA, out, dinv, b2, N);
  }

  // 6) layer-2 edge scatter into d_out (already seeded; no final pass needed)
  gcn_scatter_kernel<32, 8><<<((long long)E * 8 + TB - 1) / TB, TB, 0, stream>>>(
      bufA, dinv, src, dst, out, E);
}